// MultiHeadSliddingWindowAttention_15839839388319
// MI455X (gfx1250) — compile-verified
//
#include <hip/hip_runtime.h>
#include <hip/hip_bf16.h>
#include <math.h>

// ---------------------------------------------------------------------------
// MultiHeadSlidingWindowAttention (faithful to the reference's column-scatter
// softmax):  out_i = (e^{s0} V0 + e^{s1} V1 + e^{s2} V2 + S) / (e^{s0}+e^{s1}+e^{s2}+N-3)
// with S = sum_{j>=3} V_j  (per batch/head), s* = 3 banded QK dots per row.
// Heavy work = 4 GEMMs [4096x512]x[512x512] done with V_WMMA_F32_16X16X4_F32
// (exact fp32 matrix path on CDNA5). Each wave computes a 32(M) x 64(N) tile:
// 6 global_load_b64 feed 8 WMMAs per K-step (load:WMMA = 0.75).
// ---------------------------------------------------------------------------

#define EMBED 512
#define NSEQ  2048
#define BATCH 2
#define NHEAD 8
#define HDIM  64

typedef __attribute__((ext_vector_type(2))) float v2f;
typedef __attribute__((ext_vector_type(8))) float v8f;

__device__ __forceinline__ v8f wmma_f32(v2f a, v2f b, v8f c) {
  // 8 args: (neg_a, A, neg_b, B, c_mod, C, reuse_a, reuse_b)
  return __builtin_amdgcn_wmma_f32_16x16x4_f32(false, a, false, b, (short)0, c,
                                               false, false);
}

// Y[m,n] = sum_k X[m,k] * W[n,k] + bias[n].
// Wave computes a 32(M) x 64(N) tile; block = 8 waves = 32(M) x 512(N).
// grid.x = M/32.  K = EMBED = 512, N = 512, no divergence -> EXEC all ones.
//
// V_WMMA_F32_16X16X4_F32 fragment layout (wave32):
//   A: lane l holds A[l%16][2*(l>>4)+v], v=0,1   (float2)
//   B: lane l holds B[2*(l>>4)+v][l%16], v=0,1   (float2)
//   D: lane l, vgpr r holds D[r + 8*(l>>4)][l%16]
// Because W is [out,in] row-major, B(k,n)=W[n][k] -> same load pattern as A.
__device__ __forceinline__ void gemm_xwT_tile(const float* __restrict__ X,
                                              const float* __restrict__ W,
                                              const float* __restrict__ bias,
                                              float* __restrict__ Y) {
  const int lane = threadIdx.x & 31;
  const int wave = threadIdx.x >> 5;        // 0..7 -> n-tile group of 64 cols
  const int m0   = blockIdx.x * 32;
  const int n0   = wave * 64;
  const int rl   = lane & 15;               // M for A-frag, N for B/D-frag
  const int kk   = (lane >> 4) << 1;        // k sub-offset {0,2}

  const float* xp = X + (size_t)(m0 + rl) * EMBED + kk;
  const float* wp = W + (size_t)(n0 + rl) * EMBED + kk;

  v8f acc[4][2] = {};

#pragma unroll 2
  for (int k = 0; k < EMBED; k += 4) {
    v2f a0 = *(const v2f*)(xp + k);                            // rows m0..m0+15
    v2f a1 = *(const v2f*)(xp + (size_t)16 * EMBED + k);       // rows m0+16..m0+31
#pragma unroll
    for (int t = 0; t < 4; ++t) {
      v2f bfrag = *(const v2f*)(wp + (size_t)(t * 16) * EMBED + k);
      acc[t][0] = wmma_f32(a0, bfrag, acc[t][0]);
      acc[t][1] = wmma_f32(a1, bfrag, acc[t][1]);
    }
  }

#pragma unroll
  for (int mt = 0; mt < 2; ++mt) {
    const int mrow = m0 + mt * 16 + ((lane >> 4) << 3);
#pragma unroll
    for (int t = 0; t < 4; ++t) {
      const int col = n0 + t * 16 + rl;
      const float bv = bias[col];
      float* yp = Y + (size_t)mrow * EMBED + col;
#pragma unroll
      for (int r = 0; r < 8; ++r)
        yp[(size_t)r * EMBED] = acc[t][mt][r] + bv;
    }
  }
}

__global__ void __launch_bounds__(256)
qkv_kernel(const float* __restrict__ x,
           const float* __restrict__ Wq, const float* __restrict__ bq,
           const float* __restrict__ Wk, const float* __restrict__ bk,
           const float* __restrict__ Wv, const float* __restrict__ bv,
           float* __restrict__ Q, float* __restrict__ K, float* __restrict__ V) {
  const float* W; const float* b; float* Y;
  if (blockIdx.y == 0)      { W = Wq; b = bq; Y = Q; }
  else if (blockIdx.y == 1) { W = Wk; b = bk; Y = K; }
  else                      { W = Wv; b = bv; Y = V; }
  gemm_xwT_tile(x, W, b, Y);
}

__global__ void __launch_bounds__(256)
proj_kernel(const float* __restrict__ A, const float* __restrict__ Wo,
            const float* __restrict__ bo, float* __restrict__ out) {
  gemm_xwT_tile(A, Wo, bo, out);
}

// S[bh][d] = sum_{j=3..N-1} V[b, j, h*64+d].  One block per (b,h).
__global__ void __launch_bounds__(256)
vsum_kernel(const float* __restrict__ V, float* __restrict__ S) {
  const int bh = blockIdx.x;
  const int b = bh >> 3, h = bh & 7;
  const int t = threadIdx.x;
  const int d = t & 63;
  const int chunk = t >> 6;                 // 0..3
  float acc = 0.0f;
  for (int j = 3 + chunk; j < NSEQ; j += 4)
    acc += V[((size_t)b * NSEQ + j) * EMBED + h * HDIM + d];
  __shared__ float red[256];
  red[t] = acc;
  __syncthreads();
  if (chunk == 0)
    S[bh * HDIM + d] = red[d] + red[64 + d] + red[128 + d] + red[192 + d];
}

// One wave32 per (b,h,i): 3 banded 64-dim dots + stable softmax combine.
__global__ void __launch_bounds__(256)
attn_kernel(const float* __restrict__ Q, const float* __restrict__ K,
            const float* __restrict__ V, const float* __restrict__ S,
            float* __restrict__ OH) {
  const int gwave = (blockIdx.x * blockDim.x + threadIdx.x) >> 5;
  const int lane  = threadIdx.x & 31;
  const int i = gwave & (NSEQ - 1);
  const int h = (gwave >> 11) & (NHEAD - 1);
  const int b = gwave >> 14;

  const size_t rowQ = ((size_t)b * NSEQ + i) * EMBED + h * HDIM;
  const float q0 = Q[rowQ + lane];
  const float q1 = Q[rowQ + lane + 32];

  auto dotK = [&](int j) -> float {
    const size_t rk = ((size_t)b * NSEQ + j) * EMBED + h * HDIM;
    float p = q0 * K[rk + lane] + q1 * K[rk + lane + 32];
#pragma unroll
    for (int m = 16; m >= 1; m >>= 1) p += __shfl_xor(p, m, 32);
    return p;
  };

  const float s1 = dotK(i);
  const float s0 = (i >= 1)        ? dotK(i - 1) : 0.0f;   // wave-uniform branch
  const float s2 = (i <= NSEQ - 2) ? dotK(i + 1) : 0.0f;

  const float mx = fmaxf(fmaxf(s0, s1), fmaxf(s2, 0.0f));
  const float e0 = expf(s0 - mx);
  const float e1 = expf(s1 - mx);
  const float e2 = expf(s2 - mx);
  const float z  = expf(-mx);
  const float inv = 1.0f / (e0 + e1 + e2 + (float)(NSEQ - 3) * z);

  const size_t v0 = ((size_t)b * NSEQ) * EMBED + h * HDIM;  // V rows 0,1,2
  const float* sp = S + (b * NHEAD + h) * HDIM;
#pragma unroll
  for (int t = 0; t < 2; ++t) {
    const int d = lane + t * 32;
    const float num = e0 * V[v0 + d] + e1 * V[v0 + EMBED + d] +
                      e2 * V[v0 + 2 * EMBED + d] + z * sp[d];
    OH[rowQ + d] = num * inv;
  }
}

extern "C" void kernel_launch(void* const* d_in, const int* in_sizes, int n_in,
                              void* d_out, int out_size, void* d_ws, size_t ws_size,
                              hipStream_t stream) {
  const float* x  = (const float*)d_in[0];
  const float* Wq = (const float*)d_in[1];
  const float* bq = (const float*)d_in[2];
  const float* Wk = (const float*)d_in[3];
  const float* bk = (const float*)d_in[4];
  const float* Wv = (const float*)d_in[5];
  const float* bv = (const float*)d_in[6];
  const float* Wo = (const float*)d_in[7];
  const float* bo = (const float*)d_in[8];

  const size_t rows = (size_t)BATCH * NSEQ;       // 4096
  float* ws = (float*)d_ws;
  float* Q  = ws;
  float* K  = Q + rows * EMBED;
  float* V  = K + rows * EMBED;
  float* OH = V + rows * EMBED;
  float* S  = OH + rows * EMBED;                  // 16*64 floats

  // QKV projections (fused, one launch): grid = (M/32, 3)
  qkv_kernel<<<dim3(rows / 32, 3), 256, 0, stream>>>(x, Wq, bq, Wk, bk, Wv, bv,
                                                     Q, K, V);
  // Uniform V tail sums per (b,h)
  vsum_kernel<<<BATCH * NHEAD, 256, 0, stream>>>(V, S);
  // Banded-score softmax combine: one wave per (b,h,i)
  attn_kernel<<<(BATCH * NHEAD * NSEQ) / 8, 256, 0, stream>>>(Q, K, V, S, OH);
  // Output projection straight into d_out
  proj_kernel<<<dim3(rows / 32, 1), 256, 0, stream>>>(OH, Wo, bo, (float*)d_out);
}